// UnfoldTransposeNd_15315853378113
// MI455X (gfx1250) — compile-verified
//
#include <hip/hip_runtime.h>
#include <hip/hip_bf16.h>

// out[b, c*9 + (kh*3+kw), oh, ow] = x[b, c, (oh+1-kh)/2, (ow+1-kw)/2] when
// integral & in-range, else 0.  B=16, C=64, H=W=64, Hout=Wout=128.
//
// Pure data movement: ~604MB of output writes vs 16.7MB of (L2-resident)
// input -> write-bandwidth bound (~26us at 23.3 TB/s).  No WMMA: any matmul
// formulation adds 64 MACs/element of pure overhead.  Instead:
//   - TDM tensor_load_to_lds stages the 17 input rows a block needs (<=4.3KB),
//     serving the 9x reuse of each input row from LDS.
//   - One wave per 128-float output row, one float4 NT store per lane
//     (512B/wave coalesced, streaming past L2 since output >> L2).

typedef unsigned int u32;
typedef unsigned int v4u __attribute__((ext_vector_type(4)));
typedef int          v8i __attribute__((ext_vector_type(8)));
typedef int          v4i __attribute__((ext_vector_type(4)));
typedef float        v4f __attribute__((ext_vector_type(4)));

__global__ __launch_bounds__(256) void unfold_tr2d_kernel(
    const float* __restrict__ x, float* __restrict__ out) {
  // One shared array -> LDS offset 0 (used as the TDM D# lds_addr).
  __shared__ float lds[17 * 64];

  const u32 block = blockIdx.x;        // (b*64 + c) * 4 + q
  const u32 q     = block & 3;         // quarter of the oh range
  const u32 bc    = block >> 2;        // b*64 + c
  const u32 ih0   = q * 16;            // first input row needed for this tile
  const u32 rows  = (ih0 + 17 <= 64) ? 17u : (64u - ih0);

  const float* src = x + ((size_t)bc * 64u + ih0) * 64u;  // x[b, c, ih0, 0]

#if __has_builtin(__builtin_amdgcn_tensor_load_to_lds)
  if (threadIdx.x < 32) {  // one wave issues the TDM op (EXEC ignored)
    const unsigned long long ga = (unsigned long long)(const void*)src;

    // D# group 0: count=1, lds_addr=0, 57-bit global_addr, type=2.
    v4u g0;
    g0.x = 1u;                                   // count = 1 valid descriptor
    g0.y = 0u;                                   // lds_addr (bytes) = 0
    g0.z = (u32)ga;                              // global_addr[31:0]
    g0.w = ((u32)(ga >> 32) & 0x01FFFFFFu)       // global_addr[56:32]
           | 0x80000000u;                        // type = 2 ("image")

    // D# group 1: data_size=4B, tensor 64 x rows, tile 64 x rows,
    // dim0 stride 64, dim1 stride 4096 (elements).
    v8i g1;
    g1.s0 = (int)(2u << 16);                     // data_size = 2 (4 bytes)
    g1.s1 = (int)(64u << 16);                    // tensor_dim0 = 64 (lo16)
    g1.s2 = (int)(rows << 16);                   // dim0 hi16=0 | tensor_dim1 lo16
    g1.s3 = (int)(64u << 16);                    // dim1 hi16=0 | tile_dim0 = 64
    g1.s4 = (int)rows;                           // tile_dim1 = rows, tile_dim2 = 0
    g1.s5 = 64;                                  // tensor_dim0_stride lo32
    g1.s6 = (int)(4096u << 16);                  // stride0 hi=0 | stride1 lo16
    g1.s7 = 0;                                   // stride1 hi bits = 0

    // Groups 2/3: benign 1-extents for unused dims.
    v4i g2; g2.x = 1; g2.y = 1; g2.z = 0; g2.w = 0;         // tensor_dim2/3=1
    v4i g3; g3.x = 0; g3.y = (int)(1u << 16); g3.z = 0; g3.w = 0; // tensor_dim4=1

    // clang-23 / therock-10.0 form: 6 args with an extra int32x8 group.
    v8i g4; g4.s0 = 0; g4.s1 = 0; g4.s2 = 0; g4.s3 = 0;
            g4.s4 = 0; g4.s5 = 0; g4.s6 = 0; g4.s7 = 0;

    __builtin_amdgcn_tensor_load_to_lds(g0, g1, g2, g3, g4, 0);
    __builtin_amdgcn_s_wait_tensorcnt(0);
  }
#else
  // Fallback: cooperative global -> LDS copy.
  for (u32 i = threadIdx.x; i < rows * 64u; i += 256u) lds[i] = src[i];
#endif
  // The TDM engine wrote LDS through the descriptor, which the compiler
  // can't see (round 2: it folded the ds reads to undef -> ds=0).  Escape
  // the LDS pointer into a memory-clobbering asm so the reads are real.
  asm volatile("" : : "v"(&lds[0]) : "memory");
  __syncthreads();

  const u32 b = bc >> 6;
  const u32 c = bc & 63u;
  float* outbc = out + (size_t)(b * 576u + c * 9u) * 16384u;

  const u32 tid = threadIdx.x;
  // 9 planes * 32 oh rows = 288 rows of 128 floats; 32 float4 units per row.
  // u = iter*256 + tid -> each wave owns one full row per iteration.
#pragma unroll 4
  for (u32 iter = 0; iter < 36u; ++iter) {
    const u32 u  = iter * 256u + tid;
    const u32 r  = u >> 5;               // row id in [0, 288)
    const u32 l  = u & 31u;              // lane's float4 slot in the row
    const u32 k  = r >> 5;               // plane kh*3+kw in [0, 9)
    const u32 oh = q * 32u + (r & 31u);
    const u32 kh = k / 3u;
    const u32 kw = k - kh * 3u;

    v4f val = {0.f, 0.f, 0.f, 0.f};
    const int t = (int)oh + 1 - (int)kh;       // oh + PAD - kh*DIL
    if (!(t & 1)) {                            // parity: row is populated
      const u32 ih = (u32)(t >> 1);
      if (ih < 64u) {
        const u32 lrow = ih - ih0;             // in [0, rows)
        const u32 iw0  = 2u * l + (kw == 0u ? 1u : 0u);
        const float v0 = lds[lrow * 64u + iw0];
        const float v1 = (iw0 + 1u < 64u) ? lds[lrow * 64u + iw0 + 1u] : 0.f;
        if (((kw + 1u) & 1u) == 0u) { val.x = v0; val.z = v1; }  // kw == 1
        else                        { val.y = v0; val.w = v1; }  // kw == 0,2
      }
    }
    // Streaming 128-bit store: out[b, c*9+k, oh, 4l .. 4l+3]
    __builtin_nontemporal_store(
        val, (v4f*)(outbc + (size_t)k * 16384u + (size_t)oh * 128u + 4u * l));
  }
}

extern "C" void kernel_launch(void* const* d_in, const int* in_sizes, int n_in,
                              void* d_out, int out_size, void* d_ws, size_t ws_size,
                              hipStream_t stream) {
  (void)in_sizes; (void)n_in; (void)d_ws; (void)ws_size; (void)out_size;
  const float* x = (const float*)d_in[0];
  float* out = (float*)d_out;
  // 16 (b) * 64 (c) * 4 (oh quarters) = 4096 blocks, 256 threads each.
  unfold_tr2d_kernel<<<4096, 256, 0, stream>>>(x, out);
}